// ConcatLayer_37589553774933
// MI455X (gfx1250) — compile-verified
//
#include <hip/hip_runtime.h>

// ---------------------------------------------------------------------------
// Row-wise "tri-choice" op: in (N,9) fp32 -> out (N,3) fp32.
// Memory-bound: 402 MB total traffic -> ~17us at 23.3 TB/s on MI455X.
// Strategy: stream 9216B block chunks into LDS with gfx1250 async-to-LDS
// B128 transfers (perfectly coalesced despite 36B rows), compute per-row in
// registers, stage 3-float results in LDS, and write back as async B128.
// ---------------------------------------------------------------------------

#define THREADS 256
#define ROWS_PER_BLOCK 256
#define IN_FLOATS  (ROWS_PER_BLOCK * 9)   // 2304 floats = 576 float4
#define OUT_FLOATS (ROWS_PER_BLOCK * 3)   // 768 floats = 192 float4
#define IN_F4  (IN_FLOATS / 4)            // 576
#define OUT_F4 (OUT_FLOATS / 4)           // 192

typedef int v4i __attribute__((ext_vector_type(4)));

#if defined(__gfx1250__) && __has_builtin(__builtin_amdgcn_global_load_async_to_lds_b128)
#define HAVE_ASYNC_LOAD 1
#else
#define HAVE_ASYNC_LOAD 0
#endif

#if defined(__gfx1250__) && __has_builtin(__builtin_amdgcn_global_store_async_from_lds_b128)
#define HAVE_ASYNC_STORE 1
#else
#define HAVE_ASYNC_STORE 0
#endif

__device__ __forceinline__ int max_index3(float a, float b, float c) {
  // Mirrors reference _get_max_index: 1-first_argmax if unique max else 0.
  const float mx = fmaxf(a, fmaxf(b, c));
  const int n = (int)(a == mx) + (int)(b == mx) + (int)(c == mx);
  const int first = (a == mx) ? 0 : ((b == mx) ? 1 : 2);
  return (n == 1) ? (1 - first) : 0;
}

__global__ __launch_bounds__(THREADS)
void tri_choice_kernel(const float* __restrict__ in, float* __restrict__ out,
                       int n_rows) {
  __shared__ float sIn[IN_FLOATS];
  __shared__ float sOut[OUT_FLOATS];

  const int tid = threadIdx.x;
  const int row0 = blockIdx.x * ROWS_PER_BLOCK;
  const int rows = min(ROWS_PER_BLOCK, n_rows - row0);
  const size_t inBase = (size_t)row0 * 9;
  const size_t outBase = (size_t)row0 * 3;

  // ---- Stage input: global -> LDS, 128-bit coalesced ----
  if (rows == ROWS_PER_BLOCK) {
    v4i* g4 = (v4i*)(in + inBase);  // 9216B chunk, 16B aligned
    v4i* s4 = (v4i*)sIn;
#if HAVE_ASYNC_LOAD
    for (int k = tid; k < IN_F4; k += THREADS) {
      __builtin_amdgcn_global_load_async_to_lds_b128(g4 + k, s4 + k,
                                                     /*offset=*/0, /*cpol=*/0);
    }
    asm volatile("s_wait_asynccnt 0" ::: "memory");
#else
    for (int k = tid; k < IN_F4; k += THREADS) s4[k] = g4[k];
#endif
  } else {
    for (int k = tid; k < rows * 9; k += THREADS) sIn[k] = in[inBase + k];
  }
  __syncthreads();

  // ---- Per-row compute (LDS stride-9 dwords: odd stride -> bank-conflict-free) ----
  if (tid < rows) {
    const float* r = &sIn[tid * 9];
    const float u0 = r[0], u1 = r[1], u2 = r[2];
    const float n0 = r[3], n1 = r[4], n2 = r[5];
    const float d0 = r[6], d1 = r[7], d2 = r[8];

    const int mu = max_index3(u0, u1, u2);
    const int mn = max_index3(n0, n1, n2);
    const int md = max_index3(d0, d1, d2);

    const int amn = (mn < 0) ? -mn : mn;
    const int calc = amn * (mu + md + mn);          // integer in [-3,3]
    const int sgn = (calc > 0) - (calc < 0);

    const float wu = (sgn == mu) ? 1.0f : 0.0f;
    const float wn = (sgn == mn) ? 1.0f : 0.0f;
    const float wd = (sgn == md) ? 1.0f : 0.0f;

    const int idx = (calc == 0) ? 1 : ((calc == 1) ? 0 : 2);
    const float pu = ((idx == 0) ? u0 : (idx == 1) ? u1 : u2) * wu;
    const float pn = ((idx == 0) ? n0 : (idx == 1) ? n1 : n2) * wn;
    const float pd = ((idx == 0) ? d0 : (idx == 1) ? d1 : d2) * wd;

    // jnp.argmax: first occurrence of the maximum.
    const float pmax = fmaxf(pu, fmaxf(pn, pd));
    const int choice = (pu == pmax) ? 0 : ((pn == pmax) ? 1 : 2);

    float o0, o1, o2;
    if (choice == 0)      { o0 = u0 * wu; o1 = u1 * wu; o2 = u2 * wu; }
    else if (choice == 1) { o0 = n0 * wn; o1 = n1 * wn; o2 = n2 * wn; }
    else                  { o0 = d0 * wd; o1 = d1 * wd; o2 = d2 * wd; }

    sOut[tid * 3 + 0] = o0;
    sOut[tid * 3 + 1] = o1;
    sOut[tid * 3 + 2] = o2;
  }
  __syncthreads();

  // ---- Stage output: LDS -> global, 128-bit coalesced ----
  if (rows == ROWS_PER_BLOCK) {
    v4i* go4 = (v4i*)(out + outBase);  // 3072B chunk, 16B aligned
    v4i* so4 = (v4i*)sOut;
#if HAVE_ASYNC_STORE
    if (tid < OUT_F4) {
      __builtin_amdgcn_global_store_async_from_lds_b128(go4 + tid, so4 + tid,
                                                        /*offset=*/0, /*cpol=*/0);
    }
    // s_endpgm performs an implicit wait-idle; no explicit asynccnt wait needed.
#else
    if (tid < OUT_F4) go4[tid] = so4[tid];
#endif
  } else {
    for (int k = tid; k < rows * 3; k += THREADS) out[outBase + k] = sOut[k];
  }
}

extern "C" void kernel_launch(void* const* d_in, const int* in_sizes, int n_in,
                              void* d_out, int out_size, void* d_ws, size_t ws_size,
                              hipStream_t stream) {
  (void)n_in; (void)out_size; (void)d_ws; (void)ws_size;
  const float* in = (const float*)d_in[0];
  float* out = (float*)d_out;
  const int n_rows = in_sizes[0] / 9;
  const int blocks = (n_rows + ROWS_PER_BLOCK - 1) / ROWS_PER_BLOCK;
  tri_choice_kernel<<<blocks, THREADS, 0, stream>>>(in, out, n_rows);
}